// CalcImpute_11879879543658
// MI455X (gfx1250) — compile-verified
//
#include <hip/hip_runtime.h>
#include <hip/hip_bf16.h>

// KNN-impute (weights='distance', k=16) for 8192 x 32768 f32 distance matrix.
// Memory-bound: 1 GiB single-pass read -> ~46us floor @ 23.3 TB/s.
// One wave32 per row; async global->LDS double-buffered tiles (CDNA5 path:
// global_load_async_to_lds_b128 + s_wait_asynccnt); wave-min threshold
// filter keeps top-16 maintenance far below the memory roofline.

#define N_DON_C   32768
#define K_NEIGH   16
#define TILE_F    1024                  // floats per tile (4 KB)
#define N_TILES   (N_DON_C / TILE_F)    // 32
#define WAVES_PB  8
#define THREADS_PB (WAVES_PB * 32)
#define FLT_MAX_C 3.402823466e+38f
#define NAN_BIG_C 1.0e10f

#if defined(__has_builtin)
#  if __has_builtin(__builtin_amdgcn_global_load_async_to_lds_b128)
#    define HAVE_ASYNC_B 1
#  else
#    define HAVE_ASYNC_B 0
#  endif
#  if __has_builtin(__builtin_amdgcn_s_wait_asynccnt)
#    define HAVE_WAITA_B 1
#  else
#    define HAVE_WAITA_B 0
#  endif
#  if __has_builtin(__builtin_amdgcn_s_wait_dscnt)
#    define HAVE_WAITD_B 1
#  else
#    define HAVE_WAITD_B 0
#  endif
#else
#  define HAVE_ASYNC_B 0
#  define HAVE_WAITA_B 0
#  define HAVE_WAITD_B 0
#endif

typedef float v4f __attribute__((ext_vector_type(4)));
typedef int   v4i __attribute__((ext_vector_type(4)));

#define CFENCE() asm volatile("" ::: "memory")

#if HAVE_WAITA_B
#  define WAIT_ASYNCCNT(n) __builtin_amdgcn_s_wait_asynccnt(n)
#else
#  define WAIT_ASYNCCNT(n) asm volatile("s_wait_asynccnt %0" :: "i"(n) : "memory")
#endif

#if HAVE_WAITD_B
#  define WAIT_DSCNT(n) __builtin_amdgcn_s_wait_dscnt(n)
#else
#  define WAIT_DSCNT(n) asm volatile("s_wait_dscnt %0" :: "i"(n) : "memory")
#endif

__device__ __forceinline__ void async_b128(const float* gsrc, float* ldst)
{
#if HAVE_ASYNC_B
    __builtin_amdgcn_global_load_async_to_lds_b128(
        (__attribute__((address_space(1))) v4i*)gsrc,
        (__attribute__((address_space(3))) v4i*)ldst,
        /*imm offset*/0, /*cpol*/0);
#else
    asm volatile("global_load_async_to_lds_b128 %0, %1, off"
                 :: "v"((unsigned)(unsigned long long)ldst),
                    "v"((unsigned long long)gsrc)
                 : "memory");
#endif
}

__device__ __forceinline__ bool is_inf_f32(float w)
{
    return (__float_as_uint(w) & 0x7FFFFFFFu) == 0x7F800000u;
}

__global__ __launch_bounds__(THREADS_PB)
void knn_impute16(const float* __restrict__ dist,
                  const float* __restrict__ fitX,
                  const int*   __restrict__ maskX,
                  float*       __restrict__ out,
                  int nrows)
{
    __shared__ float stage[WAVES_PB][2][TILE_F];   // 64 KB

    const int lane = threadIdx.x & 31;
    const int wv   = threadIdx.x >> 5;
    const int row  = blockIdx.x * WAVES_PB + wv;
    if (row >= nrows) return;                      // wave-uniform

    const float* rowp = dist + (size_t)row * N_DON_C;

    // per-lane running top-16 (unsorted) + current max of the 16
    float bv[16]; int bi[16];
#pragma unroll
    for (int j = 0; j < 16; ++j) { bv[j] = FLT_MAX_C; bi[j] = 0x7fffffff; }
    float curMax = FLT_MAX_C;   // max of this lane's 16
    float T      = FLT_MAX_C;   // wave-wide upper bound on row's 16th smallest
    float thr    = FLT_MAX_C;   // min(curMax, T): skip anything >= thr

    // ---- prologue: async-stage tile 0 (8 x b128 per wave = 4 KB) ----
    {
        const float* g = rowp + lane * 4;
        float*       l = &stage[wv][0][lane * 4];
#pragma unroll
        for (int q = 0; q < 8; ++q) async_b128(g + q * 128, l + q * 128);
    }

    for (int t = 0; t < N_TILES; ++t) {
        if (t + 1 < N_TILES) {
            CFENCE();
            WAIT_DSCNT(0);   // prior tile's ds_loads drained before buffer reuse
            const float* g = rowp + (t + 1) * TILE_F + lane * 4;
            float*       l = &stage[wv][(t + 1) & 1][lane * 4];
#pragma unroll
            for (int q = 0; q < 8; ++q) async_b128(g + q * 128, l + q * 128);
            WAIT_ASYNCCNT(8);   // oldest 8 (tile t) complete; tile t+1 in flight
        } else {
            WAIT_ASYNCCNT(0);
        }
        CFENCE();

        const float* buf = stage[wv][t & 1];
        const int    gb  = t * TILE_F;
#pragma unroll
        for (int q = 0; q < 8; ++q) {
            v4f x = *(const v4f*)&buf[(q * 32 + lane) * 4];
            const int e0 = gb + (q * 32 + lane) * 4;
#pragma unroll
            for (int c = 0; c < 4; ++c) {
                float v = x[c];
                v = (v != v) ? NAN_BIG_C : v;       // nan_to_num for selection
                if (v < thr) {
                    // replace the slot holding curMax (first match)
                    bool done = false;
#pragma unroll
                    for (int j = 0; j < 16; ++j) {
                        bool hit = (!done) && (bv[j] == curMax);
                        bv[j] = hit ? v        : bv[j];
                        bi[j] = hit ? (e0 + c) : bi[j];
                        done  = done || hit;
                    }
                    float m = bv[0];
#pragma unroll
                    for (int j = 1; j < 16; ++j) m = fmaxf(m, bv[j]);
                    curMax = m;
                    thr = fminf(curMax, T);
                }
            }
        }
        // refresh wave-wide filter (monotone non-increasing => stale T is safe)
        float tm = curMax;
#pragma unroll
        for (int off = 16; off; off >>= 1) tm = fminf(tm, __shfl_xor(tm, off, 32));
        T   = tm;
        thr = fminf(curMax, T);
    }

    // ---- merge: 16 rounds of (lane-local min) + (wave32 shuffle arg-min) ----
    int myIdx = 0;
#pragma unroll 1
    for (int r = 0; r < 16; ++r) {
        float mv = bv[0]; int mi = bi[0]; int ms = 0;
#pragma unroll
        for (int j = 1; j < 16; ++j) {
            bool b = (bv[j] < mv) || ((bv[j] == mv) && (bi[j] < mi));
            mv = b ? bv[j] : mv; mi = b ? bi[j] : mi; ms = b ? j : ms;
        }
        float rv = mv; int ri = mi;
#pragma unroll
        for (int off = 16; off; off >>= 1) {
            float ov = __shfl_xor(rv, off, 32);
            int   oi = __shfl_xor(ri, off, 32);
            bool  b  = (ov < rv) || ((ov == rv) && (oi < ri));
            rv = b ? ov : rv; ri = b ? oi : ri;
        }
        bool winner = (mv == rv) && (mi == ri);   // donor indices are unique
        if (winner) {
#pragma unroll
            for (int j = 0; j < 16; ++j) {
                bool h = (j == ms);
                bv[j] = h ? FLT_MAX_C : bv[j];
                bi[j] = h ? 0x7fffffff : bi[j];
            }
        }
        if (lane == r) myIdx = ri;
    }

    // ---- epilogue: exact reference weighting semantics ----
    const bool act = (lane < K_NEIGH);
    float d  = act ? rowp[myIdx]  : 1.0f;   // ORIGINAL distance (may be 0 / NaN)
    float fx = act ? fitX[myIdx]  : 0.0f;
    int   mk = act ? maskX[myIdx] : 1;

    float w    = 1.0f / d;
    bool  winf = act && is_inf_f32(w);
    unsigned long long bal = __ballot(winf ? 1 : 0);
    if (bal != 0ull) w = winf ? 1.0f : 0.0f;   // inf-row: indicator weights
    if (w != w) w = 0.0f;                       // NaN weights -> 0

    float nw  = act ? (float)(1 - mk) * w : 0.0f;
    float num = fx * nw;
#pragma unroll
    for (int off = 16; off; off >>= 1) {
        nw  += __shfl_xor(nw,  off, 32);
        num += __shfl_xor(num, off, 32);
    }
    if (lane == 0) {
        float div = (nw == 0.0f) ? 1.0f : nw;
        out[row] = num / div;
    }
}

extern "C" void kernel_launch(void* const* d_in, const int* in_sizes, int n_in,
                              void* d_out, int out_size, void* d_ws, size_t ws_size,
                              hipStream_t stream)
{
    (void)in_sizes; (void)n_in; (void)d_ws; (void)ws_size;
    const float* dist  = (const float*)d_in[0];
    // d_in[1] = n_neighbors scalar; reference uses 16 -> specialized at compile time
    const float* fitX  = (const float*)d_in[2];
    const int*   maskX = (const int*)d_in[3];
    float*       out   = (float*)d_out;

    const int nrows  = out_size;                       // 8192
    const int blocks = (nrows + WAVES_PB - 1) / WAVES_PB;
    knn_impute16<<<blocks, THREADS_PB, 0, stream>>>(dist, fitX, maskX, out, nrows);
}